// CoordinateDescent_32375463478019
// MI455X (gfx1250) — compile-verified
//
#include <hip/hip_runtime.h>

typedef float v2f __attribute__((ext_vector_type(2)));
typedef float v8f __attribute__((ext_vector_type(8)));

#define EPS 1e-08f
#define R 32
#define BATCH 4
#define MDIM 4096
#define NDIM 4096

// ---------------------------------------------------------------------------
// Gram: bmat[batch] = w^T w, w is [Kdim, R] row-major per batch.
// grid = (BATCH, R), block = 256. blockIdx.y = row i of the Gram matrix.
// ---------------------------------------------------------------------------
__global__ __launch_bounds__(256) void gram_kernel(const float* __restrict__ w,
                                                   float* __restrict__ bmat,
                                                   int Kdim) {
    const int batch = blockIdx.x;
    const int i = blockIdx.y;
    const float* wb = w + (size_t)batch * Kdim * R;
    const int j = threadIdx.x & 31;
    const int slice = threadIdx.x >> 5;

    float s = 0.f;
    for (int n = slice; n < Kdim; n += 8) {
        const float* row = wb + (size_t)n * R;
        s += row[i] * row[j];
    }
    __shared__ float red[256];
    red[threadIdx.x] = s;
    __syncthreads();
    if (slice == 0) {
        float t = 0.f;
        #pragma unroll
        for (int q = 0; q < 8; ++q) t += red[q * 32 + j];
        bmat[((size_t)batch * R + i) * R + j] = t;
    }
}

// ---------------------------------------------------------------------------
// GEMM via WMMA f32 16x16x4:  out[i, r] = sum_k A(i,k) * w[k, r]
//   TRANS == 0 : A(i,k) = x[i * Kdim + k]   (a = x @ v)
//   TRANS == 1 : A(i,k) = x[k * Mdim + i]   (a = x^T @ u)
// Each wave owns one 16-row tile and both 16-col halves of R=32.
// grid = (Mdim/128, BATCH), block = 256 (8 waves).
// ---------------------------------------------------------------------------
template <int TRANS>
__global__ __launch_bounds__(256) void gemm_wmma_kernel(const float* __restrict__ x,
                                                        const float* __restrict__ w,
                                                        float* __restrict__ out,
                                                        int Mdim, int Kdim) {
    const int lane  = threadIdx.x & 31;
    const int wave  = threadIdx.x >> 5;
    const int laneN = lane & 15;
    const int kOff  = (lane >> 4) << 1;   // 0 for lanes 0-15, 2 for lanes 16-31

    const int mTile = (blockIdx.x * 8 + wave) * 16;
    const int batch = blockIdx.y;

    const float* xb = x + (size_t)batch * Mdim * Kdim;
    const float* wb = w + (size_t)batch * Kdim * R;
    float*       ob = out + (size_t)batch * Mdim * R;

    const int aRow = mTile + laneN;   // M index held by this lane (A layout)

    v8f acc0 = {};   // cols 0..15
    v8f acc1 = {};   // cols 16..31

    #pragma unroll 4
    for (int k = 0; k < Kdim; k += 4) {
        const int kk = k + kOff;

        v2f a;
        if (TRANS == 0) {
            const float* p = xb + (size_t)aRow * Kdim + kk;  // contiguous -> b64
            a.x = p[0];
            a.y = p[1];
        } else {
            a.x = xb[(size_t)kk * Mdim + aRow];              // coalesced across lanes
            a.y = xb[(size_t)(kk + 1) * Mdim + aRow];
        }

        const float* wr0 = wb + (size_t)kk * R;   // K row kk
        const float* wr1 = wr0 + R;               // K row kk+1
        v2f b0, b1;
        b0.x = wr0[laneN];
        b0.y = wr1[laneN];
        b1.x = wr0[laneN + 16];
        b1.y = wr1[laneN + 16];

        acc0 = __builtin_amdgcn_wmma_f32_16x16x4_f32(false, a, false, b0,
                                                     (short)0, acc0, false, false);
        acc1 = __builtin_amdgcn_wmma_f32_16x16x4_f32(false, a, false, b1,
                                                     (short)0, acc1, false, false);
    }

    // C/D layout: VGPR j -> row mTile + j (lanes 0-15) or mTile + j + 8 (lanes 16-31)
    const int rBase = mTile + ((lane >> 4) << 3);
    #pragma unroll
    for (int j = 0; j < 8; ++j) {
        float* row = ob + (size_t)(rBase + j) * R;
        row[laneN]      = acc0[j];
        row[laneN + 16] = acc1[j];
    }
}

// ---------------------------------------------------------------------------
// Coordinate-descent scan: rows independent, r-loop sequential (Gauss-Seidel).
//   for r: u[r] = (a[r] - (sum_j u[j]*b[j][r] - u[r]*b[r][r]) + eps) / (b[r][r] + eps)
// grid = (Mdim/256, BATCH), block = 256; one thread per row.
// ---------------------------------------------------------------------------
__global__ __launch_bounds__(256) void cd_scan_kernel(const float* __restrict__ a,
                                                      const float* __restrict__ bmat,
                                                      const float* __restrict__ u_in,
                                                      float* __restrict__ u_out,
                                                      int Mdim) {
    const int batch = blockIdx.y;
    const int m = blockIdx.x * 256 + threadIdx.x;

    __shared__ float sB[R * R];
    for (int t = threadIdx.x; t < R * R; t += 256)
        sB[t] = bmat[(size_t)batch * R * R + t];
    __syncthreads();

    const size_t rowOff = ((size_t)batch * Mdim + m) * R;
    float ur[R], ar[R];
    #pragma unroll
    for (int r = 0; r < R; ++r) {
        ur[r] = u_in[rowOff + r];
        ar[r] = a[rowOff + r];
    }

    #pragma unroll
    for (int r = 0; r < R; ++r) {
        const float brr = sB[r * R + r];
        float dot = 0.f;
        #pragma unroll
        for (int j = 0; j < R; ++j)
            dot = fmaf(ur[j], sB[j * R + r], dot);
        const float term2 = dot - ur[r] * brr;
        ur[r] = (ar[r] - term2 + EPS) / (brr + EPS);
    }

    #pragma unroll
    for (int r = 0; r < R; ++r)
        u_out[rowOff + r] = ur[r];
}

// ---------------------------------------------------------------------------
extern "C" void kernel_launch(void* const* d_in, const int* in_sizes, int n_in,
                              void* d_out, int out_size, void* d_ws, size_t ws_size,
                              hipStream_t stream) {
    const float* x = (const float*)d_in[0];   // [B, M, N]
    const float* u = (const float*)d_in[1];   // [B, M, R]
    const float* v = (const float*)d_in[2];   // [B, N, R]

    float* u_new = (float*)d_out;                              // [B, M, R]
    float* v_new = u_new + (size_t)BATCH * MDIM * R;           // [B, N, R]

    float* a_buf = (float*)d_ws;                               // [B, max(M,N), R]
    float* bmat  = a_buf + (size_t)BATCH * MDIM * R;           // [B, R, R]

    dim3 blk(256);

    // ---- Phase 1: u_new = update_u(x, u, v) ----
    gram_kernel<<<dim3(BATCH, R), blk, 0, stream>>>(v, bmat, NDIM);
    gemm_wmma_kernel<0><<<dim3(MDIM / 128, BATCH), blk, 0, stream>>>(x, v, a_buf, MDIM, NDIM);
    cd_scan_kernel<<<dim3(MDIM / 256, BATCH), blk, 0, stream>>>(a_buf, bmat, u, u_new, MDIM);

    // ---- Phase 2: v_new = update_u(x^T, v, u_new) ----
    gram_kernel<<<dim3(BATCH, R), blk, 0, stream>>>(u_new, bmat, MDIM);
    gemm_wmma_kernel<1><<<dim3(NDIM / 128, BATCH), blk, 0, stream>>>(x, u_new, a_buf, NDIM, MDIM);
    cd_scan_kernel<<<dim3(NDIM / 256, BATCH), blk, 0, stream>>>(a_buf, bmat, v, v_new, NDIM);
}